// ConvolutionLayer_10316511445656
// MI455X (gfx1250) — compile-verified
//
#include <hip/hip_runtime.h>
#include <math.h>

#define PI_F 3.14159265358979323846f

typedef __attribute__((ext_vector_type(2))) float v2f;
typedef __attribute__((ext_vector_type(8))) float v8f;

// ---------------------------------------------------------------------------
// Kernel 1: s_b = sum over rfft bins of u[b,:]  (closed form, no FFT)
//   s_re[b] = (L/2+1)*u[b,0] + sum_{even l>=2} u[b,l]
//   s_im[b] = -sum_{odd l} cot(pi*l/L) * u[b,l]
// One block per batch row, 256-thread LDS tree reduction.
// ---------------------------------------------------------------------------
__global__ void k_batch_sums(const float* __restrict__ u,
                             float* __restrict__ s_re,
                             float* __restrict__ s_im, int L) {
  __shared__ float rre[256];
  __shared__ float rim[256];
  const int b = blockIdx.x;
  const int t = threadIdx.x;
  const float* up = u + (size_t)b * (size_t)L;
  float are = 0.0f, aim = 0.0f;
  for (int l = t; l < L; l += 256) {
    float v = up[l];
    if (l & 1) {
      float x = PI_F * (float)l / (float)L;
      aim -= v * (cosf(x) / sinf(x));
    } else {
      are += (l == 0) ? v * (float)(L / 2 + 1) : v;
    }
  }
  rre[t] = are; rim[t] = aim;
  __syncthreads();
  for (int s = 128; s > 0; s >>= 1) {
    if (t < s) { rre[t] += rre[t + s]; rim[t] += rim[t + s]; }
    __syncthreads();
  }
  if (t == 0) { s_re[b] = rre[0]; s_im[b] = rim[0]; }
}

// ---------------------------------------------------------------------------
// Kernel 2: P[l] = sum_k lambda_k * ev[k,l]; Hilbert kernel
//   kern[d] = -(2/L)*cot(pi*d/L) for odd d, else 0
// ---------------------------------------------------------------------------
__global__ void k_p_and_kern(const float* __restrict__ ev,
                             const float* __restrict__ lam,
                             float* __restrict__ P,
                             float* __restrict__ kern, int K, int L) {
  const int l = blockIdx.x * blockDim.x + threadIdx.x;
  if (l >= L) return;
  float p = 0.0f;
  for (int k = 0; k < K; ++k)
    p = fmaf(lam[k], ev[(size_t)k * (size_t)L + l], p);
  P[l] = p;
  float kv = 0.0f;
  if (l & 1) {
    float x = PI_F * (float)l / (float)L;
    kv = (-2.0f / (float)L) * (cosf(x) / sinf(x));
  }
  kern[l] = kv;
}

// ---------------------------------------------------------------------------
// Kernel 3: circular convolution Q = kern (*) P, split over the m-axis for
// occupancy: block (x = l-tile of 256, y = m-chunk s of SPLITS) computes
//   Qpart[s*L + l] = sum_{m in chunk s, (l-m) odd} kern[(l-m) mod L] * P[m]
// kern fully resident in LDS (32KB), P streamed through a 4KB LDS tile.
// Assumes L is a power of two (8192 here).
// ---------------------------------------------------------------------------
#define PTILE 1024

__global__ void k_hilbert(const float* __restrict__ P,
                          const float* __restrict__ kern,
                          float* __restrict__ Qpart, int L, int chunk) {
  __shared__ float sK[8192];
  __shared__ float sP[PTILE];
  const int t = threadIdx.x;
  const int l = blockIdx.x * 256 + t;
  const int mBase = blockIdx.y * chunk;
  for (int i = t; i < L; i += 256) sK[i] = kern[i];
  const int mask = L - 1;
  const int j0 = (l & 1) ^ 1;   // m must have opposite parity of l
  float q = 0.0f;
  for (int mt = mBase; mt < mBase + chunk; mt += PTILE) {
    __syncthreads();
    for (int i = t; i < PTILE; i += 256) sP[i] = P[mt + i];
    // hint the next tile toward the caches (global_prefetch_b8)
    if (mt + PTILE < L) __builtin_prefetch(&P[mt + PTILE + t], 0, 3);
    __syncthreads();
#pragma unroll 4
    for (int j = j0; j < PTILE; j += 2)
      q = fmaf(sK[(l - (mt + j)) & mask], sP[j], q);
  }
  Qpart[(size_t)blockIdx.y * (size_t)L + l] = q;
}

// ---------------------------------------------------------------------------
// Kernel 4: out(B x L) = S(B x 4) * [P;Q;0;0](4 x L) via v_wmma_f32_16x16x4_f32
// One wave computes one 16x16 tile. A 16x4 f32 layout (ISA 7.12.2): lanes
// 0-15 hold K=0 (vgpr0) / K=1 (vgpr1) for row M=lane; lanes 16-31 hold the
// zero K=2/3 rows. B 4x16 mirrors this. The Qpart fold (nsplit partial sums,
// fixed order -> deterministic) is fused into the B-operand load. Masking is
// arithmetic so EXEC stays all-ones around the WMMA.
// ---------------------------------------------------------------------------
__global__ void k_rank2_wmma(const float* __restrict__ s_re,
                             const float* __restrict__ s_im,
                             const float* __restrict__ P,
                             const float* __restrict__ Qpart,
                             float* __restrict__ out, int L, int nsplit) {
  const int wave = (int)((blockIdx.x * blockDim.x + threadIdx.x) >> 5);
  const int lane = (int)(threadIdx.x & 31);
  const int colTiles = L >> 4;
  const int rt = wave / colTiles;     // batch-tile (16 rows)
  const int ct = wave % colTiles;     // length-tile (16 cols)
  const int b0 = rt << 4;
  const int l0 = ct << 4;
  const int r = lane & 15;
  const float msk = (lane < 16) ? 1.0f : 0.0f;   // zero K=2,3 rows

  float qv = 0.0f;
  for (int s = 0; s < nsplit; ++s) qv += Qpart[(size_t)s * (size_t)L + l0 + r];

  v2f a; a.x = s_re[b0 + r] * msk; a.y = s_im[b0 + r] * msk;
  v2f b; b.x = P[l0 + r]   * msk; b.y = qv            * msk;
  v8f c = {};
  c = __builtin_amdgcn_wmma_f32_16x16x4_f32(
      /*neg_a=*/false, a, /*neg_b=*/false, b,
      /*c_mod=*/(short)0, c, /*reuse_a=*/false, /*reuse_b=*/false);

  // D layout: vgpr v, lanes 0-15 -> M=v, N=lane; lanes 16-31 -> M=v+8, N=lane-16
  const int col = l0 + r;
  const int rowBase = b0 + ((lane >> 4) << 3);
#pragma unroll
  for (int v = 0; v < 8; ++v)
    out[(size_t)(rowBase + v) * (size_t)L + col] = c[v];
}

// ---------------------------------------------------------------------------
extern "C" void kernel_launch(void* const* d_in, const int* in_sizes, int n_in,
                              void* d_out, int out_size, void* d_ws, size_t ws_size,
                              hipStream_t stream) {
  const float* u   = (const float*)d_in[0];   // (B, L)
  const float* ev  = (const float*)d_in[1];   // (K, L)
  const float* lam = (const float*)d_in[2];   // (K,)
  float* out = (float*)d_out;                 // (B, L)

  const int K = in_sizes[2];
  const int L = in_sizes[1] / K;              // 8192
  const int B = in_sizes[0] / L;              // 64

  // Split factor for the m-reduction: 8-way if workspace allows (deterministic
  // per-run since ws_size is fixed), else single chunk.
  int nsplit = 8;
  size_t need = ((size_t)2 * B + (size_t)2 * L + (size_t)nsplit * L) * sizeof(float);
  if (ws_size < need) nsplit = 1;
  const int chunk = L / nsplit;

  float* ws    = (float*)d_ws;
  float* s_re  = ws;                          // B
  float* s_im  = ws + B;                      // B
  float* P     = ws + 2 * B;                  // L
  float* kern  = P + L;                       // L
  float* Qpart = kern + L;                    // nsplit * L

  k_batch_sums<<<B, 256, 0, stream>>>(u, s_re, s_im, L);
  k_p_and_kern<<<(L + 255) / 256, 256, 0, stream>>>(ev, lam, P, kern, K, L);
  k_hilbert<<<dim3(L / 256, nsplit), 256, 0, stream>>>(P, kern, Qpart, L, chunk);

  const int waves  = (B / 16) * (L / 16);     // 4 * 512 = 2048 tiles
  const int blocks = (waves * 32) / 256;      // 8 waves per block
  k_rank2_wmma<<<blocks, 256, 0, stream>>>(s_re, s_im, P, Qpart, out, L, nsplit);
}